// AdaptiveGraphConvolution_11656541241695
// MI455X (gfx1250) — compile-verified
//
#include <hip/hip_runtime.h>
#include <stdint.h>

typedef float v2f __attribute__((ext_vector_type(2)));
typedef float v8f __attribute__((ext_vector_type(8)));

// D = A(16x4) * B(4x16) + C, fp32, exact same precision as reference math.
#define WMMA_F32(a, b, c) \
  __builtin_amdgcn_wmma_f32_16x16x4_f32(false, (a), false, (b), (short)0, (c), false, false)

// ---- async global->LDS copy (CDNA5), with portable fallback ---------------
// Probe result (round 2 diagnostic): builtin exists with signature
//   void(__attribute__((address_space(1))) int* src_global,
//        __attribute__((address_space(3))) int* dst_lds, int offset, int cpol)
#if __has_builtin(__builtin_amdgcn_global_load_async_to_lds_b32)
#define AS1I(p) ((__attribute__((address_space(1))) int*)(p))
#define AS3I(p) ((__attribute__((address_space(3))) int*)(p))
#define XCOPY(dst, src) \
  __builtin_amdgcn_global_load_async_to_lds_b32(AS1I(src), AS3I(dst), 0, 0)
#if __has_builtin(__builtin_amdgcn_s_wait_asynccnt)
#define XWAIT() __builtin_amdgcn_s_wait_asynccnt(0)
#else
#define XWAIT() asm volatile("s_wait_asynccnt 0x0" ::: "memory")
#endif
#else
#define XCOPY(dst, src) (*(dst) = *(src))
#define XWAIT() ((void)0)
#endif

static constexpr int N_  = 64;
static constexpr int C_  = 64;
static constexpr int T_  = 300;
static constexpr int V_  = 25;
static constexpr int CI_ = 16;
static constexpr int S_  = 3;

static constexpr int TK1  = 4;          // T chunks for kernel1 (partial logits)
static constexpr int TC1  = T_ / TK1;   // 75 t per block
static constexpr int TPB2 = 5;          // t per block in kernel2
static constexpr int NB2T = T_ / TPB2;  // 60 chunks

// Branch-free x_t tile loader: copies the 64x25 block for one t into an LDS
// tile of row stride STRIDE (pad columns pre-zeroed once by the caller).
template <int NT, int STRIDE>
__device__ __forceinline__ void load_xt(float* __restrict__ buf,
                                        const float* __restrict__ base, int tid)
{
  constexpr int TOT  = C_ * V_;      // 1600
  constexpr int FULL = TOT / NT;
#pragma unroll
  for (int k = 0; k < FULL; ++k) {
    const int idx = tid + k * NT;
    const int c = idx / V_, v = idx - c * V_;
    XCOPY(&buf[c * STRIDE + v], &base[c * (T_ * V_) + v]);
  }
  {
    const int idx = tid + FULL * NT;
    if (idx < TOT) {
      const int c = idx / V_, v = idx - c * V_;
      XCOPY(&buf[c * STRIDE + v], &base[c * (T_ * V_) + v]);
    }
  }
}

// -------------------------------------------------------------------------
// Kernel 1: partial attention logits.
// block: 128 threads (4 waves). grid: (chunk, n, subset)
// stage A: a_t = Wa_i @ x_t, b_t = Wb_i @ x_t   (M=16, N=25->32, K=64)
// stage B: logits += a_t^T @ b_t                (M=N=25->32, K=16)
// x_t double-buffered; loads for t+1 overlap the WMMA chain of t.
// -------------------------------------------------------------------------
__global__ __launch_bounds__(128)
void k1_logits(const float* __restrict__ x,
               const float* __restrict__ Wa, const float* __restrict__ ba,
               const float* __restrict__ Wb, const float* __restrict__ bb,
               float* __restrict__ partial)
{
  __shared__ float x_lds[2][64 * 32];  // x_t, cols 25..31 pre-zeroed
  __shared__ float a_lds[16 * 33];
  __shared__ float b_lds[16 * 33];

  const int tid  = threadIdx.x;
  const int w    = tid >> 5;
  const int lane = tid & 31;
  const int mrow = lane & 15;
  const int hi   = lane >> 4;

  const int chunk = blockIdx.x;
  const int n     = blockIdx.y;
  const int i     = blockIdx.z;

  // Preload stage-A weight fragments into registers (wave 0/1 -> Wa, 2/3 -> Wb)
  const float* Wsel = ((w < 2) ? Wa : Wb) + i * CI_ * C_;
  const float* bsel = ((w < 2) ? ba : bb) + i * CI_;
  v2f wf[16];
#pragma unroll
  for (int k = 0; k < 16; ++k) {
    const int kb = 4 * k + 2 * hi;
    wf[k].x = Wsel[mrow * C_ + kb];
    wf[k].y = Wsel[mrow * C_ + kb + 1];
  }
  const int n0A = w & 1;
  float* out_l = (w < 2) ? a_lds : b_lds;
  const int m0 = w >> 1, n0 = w & 1;

  // Zero the pad columns of both x buffers once (2 * 64 rows * 7 cols)
  for (int idx = tid; idx < 2 * 64 * 7; idx += 128) {
    const int bi = idx / 448, rem = idx % 448;
    const int c = rem / 7, v = V_ + rem % 7;
    x_lds[bi][c * 32 + v] = 0.f;
  }

  const float* xn = x + (size_t)n * (C_ * T_ * V_);
  const int t0 = chunk * TC1;

  load_xt<128, 32>(x_lds[0], xn + t0 * V_, tid);   // prologue: t0 into buf0

  v8f logits = {};

  for (int tt = 0; tt < TC1; ++tt) {
    const int cur = tt & 1;
    const float* xb = x_lds[cur];

    XWAIT();
    __syncthreads();  // publish x[cur] (+ pads); all prior a/b reads done

    if (tt + 1 < TC1)
      load_xt<128, 32>(x_lds[cur ^ 1], xn + (t0 + tt + 1) * V_, tid);

    // ---- stage A ----
    v8f acc = {};
#pragma unroll
    for (int k = 0; k < 16; ++k) {
      const int kb = 4 * k + 2 * hi;
      v2f bf;
      bf.x = xb[kb * 32 + n0A * 16 + mrow];
      bf.y = xb[(kb + 1) * 32 + n0A * 16 + mrow];
      acc = WMMA_F32(wf[k], bf, acc);
    }
    {
      union { v8f v; float f[8]; } u; u.v = acc;
      const int col = n0A * 16 + mrow;
#pragma unroll
      for (int r = 0; r < 8; ++r) {
        const int row = r + hi * 8;
        const float bias = (col < V_) ? bsel[row] : 0.f;  // keep pad cols zero
        out_l[row * 33 + col] = u.f[r] + bias;
      }
    }
    __syncthreads();  // a/b tiles ready

    // ---- stage B: logits += a^T @ b ----
#pragma unroll
    for (int kk = 0; kk < 4; ++kk) {
      const int kb = 4 * kk + 2 * hi;
      v2f af, bf;
      af.x = a_lds[kb * 33 + m0 * 16 + mrow];
      af.y = a_lds[(kb + 1) * 33 + m0 * 16 + mrow];
      bf.x = b_lds[kb * 33 + n0 * 16 + mrow];
      bf.y = b_lds[(kb + 1) * 33 + n0 * 16 + mrow];
      logits = WMMA_F32(af, bf, logits);
    }
    // next iteration's top barrier protects a_lds/b_lds before rewrite
  }

  const float scale = 1.f / (float)(CI_ * T_);
  union { v8f v; float f[8]; } u; u.v = logits;
  float* dst = partial + (size_t)((i * N_ + n) * TK1 + chunk) * (V_ * V_);
#pragma unroll
  for (int r = 0; r < 8; ++r) {
    const int v1 = m0 * 16 + r + hi * 8;
    const int v2 = n0 * 16 + mrow;
    if (v1 < V_ && v2 < V_) dst[v1 * V_ + v2] = u.f[r] * scale;
  }
}

// -------------------------------------------------------------------------
// Kernel 1b: reduce chunks, column softmax (over v1), add A_static+graph_attn,
// emit zero-padded 28x32 attention tile per (i,n).
// -------------------------------------------------------------------------
__global__ __launch_bounds__(32)
void k1b_softmax(const float* __restrict__ partial,
                 const float* __restrict__ A_static,
                 const float* __restrict__ gattn,
                 float* __restrict__ attnp)
{
  const int n = blockIdx.x, i = blockIdx.y;
  const int v2 = threadIdx.x;

  float col[V_];
  float mx = -3.0e38f, s = 0.f;
  if (v2 < V_) {
#pragma unroll
    for (int v1 = 0; v1 < V_; ++v1) {
      float sum = 0.f;
#pragma unroll
      for (int c = 0; c < TK1; ++c)
        sum += partial[(size_t)((i * N_ + n) * TK1 + c) * (V_ * V_) + v1 * V_ + v2];
      col[v1] = sum;
      mx = fmaxf(mx, sum);
    }
#pragma unroll
    for (int v1 = 0; v1 < V_; ++v1) { col[v1] = __expf(col[v1] - mx); s += col[v1]; }
  }
  const float inv_s = (v2 < V_) ? 1.f / s : 0.f;

  float* dst = attnp + (size_t)(i * N_ + n) * (28 * 32);
#pragma unroll
  for (int v1 = 0; v1 < 28; ++v1) {
    float val = 0.f;
    if (v1 < V_ && v2 < V_)
      val = col[v1] * inv_s
          + A_static[(i * V_ + v1) * V_ + v2]
          + gattn[(i * V_ + v1) * V_ + v2];
    dst[v1 * 32 + v2] = val;
  }
}

// -------------------------------------------------------------------------
// Kernel 2: per (n,t): for each subset i: y = x_t @ attn_i ; h += Wg_i @ y,
// then fused bias + BN + residual + ReLU store.
// block: 256 threads (8 waves, one 16x16 output tile each). grid: (tchunk, n)
// x and y double-buffered; 1 + 3 barriers per t.
// -------------------------------------------------------------------------
__global__ __launch_bounds__(256)
void k2_main(const float* __restrict__ x, const float* __restrict__ attnp,
             const float* __restrict__ Wg, const float* __restrict__ bg,
             const float* __restrict__ gamma, const float* __restrict__ beta,
             const float* __restrict__ rmean, const float* __restrict__ rvar,
             float* __restrict__ out)
{
  __shared__ float x_lds[2][64 * 29];     // x_t, cols 25..28 pre-zeroed
  __shared__ float y_lds[2][64 * 32];     // y_t staging, parity by subset
  __shared__ float attn_lds[3 * 28 * 32]; // padded attention tiles
  __shared__ float inv_s[64];
  __shared__ float shift_s[64];

  const int tid  = threadIdx.x;
  const int w    = tid >> 5;
  const int lane = tid & 31;
  const int mrow = lane & 15;
  const int hi   = lane >> 4;
  const int m0   = w >> 1, n0 = w & 1;

  const int n     = blockIdx.y;
  const int tbase = blockIdx.x * TPB2;

  // Preamble: attn tiles + BN constants into LDS, zero x pad columns
  for (int idx = tid; idx < 3 * 28 * 32; idx += 256) {
    const int i = idx / (28 * 32), j = idx % (28 * 32);
    attn_lds[idx] = attnp[(size_t)(i * N_ + n) * (28 * 32) + j];
  }
  if (tid < C_) {
    const int o = tid;
    const float inv = gamma[o] / sqrtf(rvar[o] + 1e-5f);
    const float bg3 = bg[o] + bg[C_ + o] + bg[2 * C_ + o];
    inv_s[o]   = inv;
    shift_s[o] = (bg3 - rmean[o]) * inv + beta[o];
  }
  for (int idx = tid; idx < 2 * 64 * 4; idx += 256) {
    const int bi = idx >> 8, rem = idx & 255;
    const int c = rem >> 2, v = V_ + (rem & 3);
    x_lds[bi][c * 29 + v] = 0.f;
  }

  // Preload Wg A-fragments (per-wave output rows) into registers: 3*16 v2f
  v2f wgf[3][16];
#pragma unroll
  for (int i = 0; i < 3; ++i) {
#pragma unroll
    for (int k = 0; k < 16; ++k) {
      const int kb  = 4 * k + 2 * hi;
      const int row = m0 * 16 + mrow;
      wgf[i][k].x = Wg[(i * C_ + row) * C_ + kb];
      wgf[i][k].y = Wg[(i * C_ + row) * C_ + kb + 1];
    }
  }

  const float* xn = x + (size_t)n * (C_ * T_ * V_);
  float*       on = out + (size_t)n * (C_ * T_ * V_);

  load_xt<256, 29>(x_lds[0], xn + tbase * V_, tid);   // prologue

  for (int tt = 0; tt < TPB2; ++tt) {
    const int t   = tbase + tt;
    const int cur = tt & 1;
    const float* xb = x_lds[cur];

    XWAIT();
    __syncthreads();  // publish x[cur], attn, BN consts; prior reads retired

    if (tt + 1 < TPB2)
      load_xt<256, 29>(x_lds[cur ^ 1], xn + (t + 1) * V_, tid);

    v8f h = {};
#pragma unroll
    for (int i = 0; i < 3; ++i) {
      float* yb = y_lds[i & 1];

      // GEMM1: y_t = x_t (64x25->28) @ attn_i (25->28 x 25->32)
      v8f y = {};
#pragma unroll
      for (int kk = 0; kk < 7; ++kk) {
        const int kb = 4 * kk + 2 * hi;
        v2f af, bf;
        af.x = xb[(m0 * 16 + mrow) * 29 + kb];
        af.y = xb[(m0 * 16 + mrow) * 29 + kb + 1];
        bf.x = attn_lds[i * (28 * 32) + kb * 32 + n0 * 16 + mrow];
        bf.y = attn_lds[i * (28 * 32) + (kb + 1) * 32 + n0 * 16 + mrow];
        y = WMMA_F32(af, bf, y);
      }
      {
        union { v8f v; float f[8]; } uy; uy.v = y;
#pragma unroll
        for (int r = 0; r < 8; ++r) {
          const int row = m0 * 16 + r + hi * 8;
          const int col = n0 * 16 + mrow;
          yb[row * 32 + col] = uy.f[r];
        }
      }
      __syncthreads();  // y[i&1] ready (also retires reads of y[(i-1)&1])

      // GEMM2: h += Wg_i (64x64) @ y_t (64x25->32)
#pragma unroll
      for (int k = 0; k < 16; ++k) {
        const int kb = 4 * k + 2 * hi;
        v2f bf;
        bf.x = yb[kb * 32 + n0 * 16 + mrow];
        bf.y = yb[(kb + 1) * 32 + n0 * 16 + mrow];
        h = WMMA_F32(wgf[i][k], bf, h);
      }
    }

    // Fused epilogue: bias-sum folded into shift, BN, residual, ReLU
    union { v8f v; float f[8]; } uh; uh.v = h;
#pragma unroll
    for (int r = 0; r < 8; ++r) {
      const int o = m0 * 16 + r + hi * 8;
      const int v = n0 * 16 + mrow;
      if (v < V_) {
        const float val = uh.f[r] * inv_s[o] + shift_s[o] + xb[o * 29 + v];
        on[o * (T_ * V_) + t * V_ + v] = fmaxf(val, 0.f);
      }
    }
    // next iteration's top barrier protects x/y buffers
  }
}

// -------------------------------------------------------------------------
extern "C" void kernel_launch(void* const* d_in, const int* in_sizes, int n_in,
                              void* d_out, int out_size, void* d_ws, size_t ws_size,
                              hipStream_t stream)
{
  const float* x        = (const float*)d_in[0];
  const float* A_static = (const float*)d_in[1];
  const float* gattn    = (const float*)d_in[2];
  const float* Wg       = (const float*)d_in[3];
  const float* bg       = (const float*)d_in[4];
  const float* Wa       = (const float*)d_in[5];
  const float* ba       = (const float*)d_in[6];
  const float* Wb       = (const float*)d_in[7];
  const float* bb       = (const float*)d_in[8];
  const float* gamma    = (const float*)d_in[9];
  const float* beta     = (const float*)d_in[10];
  const float* rmean    = (const float*)d_in[11];
  const float* rvar     = (const float*)d_in[12];
  float* out = (float*)d_out;

  float* ws      = (float*)d_ws;
  float* partial = ws;                                    // 3*64*4*625 floats
  float* attnp   = ws + (size_t)S_ * N_ * TK1 * V_ * V_;  // 3*64*28*32 floats

  k1_logits  <<<dim3(TK1, N_, S_), 128, 0, stream>>>(x, Wa, ba, Wb, bb, partial);
  k1b_softmax<<<dim3(N_, S_),       32, 0, stream>>>(partial, A_static, gattn, attnp);
  k2_main    <<<dim3(NB2T, N_),    256, 0, stream>>>(x, attnp, Wg, bg, gamma, beta,
                                                     rmean, rvar, out);
}